// Decoder_78915729096805
// MI455X (gfx1250) — compile-verified
//
#include <hip/hip_runtime.h>
#include <math.h>

// Problem constants (reference: B=32, L=64, E=1024, H=128, VOCAB=32000)
#define B_   32
#define L_   64
#define E_   1024
#define H_   128
#define VOC_ 32000

typedef __attribute__((ext_vector_type(2))) float v2f;
typedef __attribute__((ext_vector_type(8))) float v8f;

#define WMMA_F32(a, b, c) \
    __builtin_amdgcn_wmma_f32_16x16x4_f32(false, (a), false, (b), (short)0, (c), false, false)

// ---------------------------------------------------------------------------
// M-blocked GEMM: one wave computes a 64x16 C block (4 accumulators) so the
// strided B fragment is reused 4x per k-step.  K fixed at 128 so the k-loop
// unrolls with pointer increments (no per-iteration 64-bit multiplies).
// A: M x 128 row-major (lda), B: 128 x N row-major (ldb), C: M x N (ldc).
// M % 64 == 0, N % 16 == 0.
// ---------------------------------------------------------------------------
__global__ void gemm_wmma_mb4_k(const float* __restrict__ A,
                                const float* __restrict__ Bm,
                                float* __restrict__ C,
                                int M, int N, int lda, int ldb, int ldc) {
    const int wid  = (blockIdx.x * blockDim.x + threadIdx.x) >> 5;
    const int lane = threadIdx.x & 31;
    const int ntiles = N >> 4;
    const int nt = wid % ntiles;
    const int mt = wid / ntiles;                 // 64-row super-tile index
    if (mt * 64 >= M) return;                    // wave-uniform guard
    const int n0 = nt << 4, m0 = mt << 6;
    const int mn = lane & 15;                    // A row / B col within tile
    const int kb = (lane < 16) ? 0 : 2;          // K sub-pair per lane half

    const float* Ar0 = A + (size_t)(m0 + mn) * lda + kb;
    const float* Ar1 = Ar0 + (size_t)16 * lda;
    const float* Ar2 = Ar0 + (size_t)32 * lda;
    const float* Ar3 = Ar0 + (size_t)48 * lda;
    const float* Bc  = Bm + (size_t)kb * ldb + n0 + mn;

    v8f c0 = {0.f,0.f,0.f,0.f,0.f,0.f,0.f,0.f};
    v8f c1 = c0, c2 = c0, c3 = c0;

#pragma unroll 4
    for (int k0 = 0; k0 < 128; k0 += 4) {
        if ((k0 & 31) == 0 && k0 + 36 <= 128) {  // uniform-branch B-panel prefetch
            __builtin_prefetch(Bc + (size_t)32 * ldb, 0, 1);
        }
        v2f b;
        b.x = Bc[0];
        b.y = Bc[ldb];
        v2f a0; a0.x = Ar0[0]; a0.y = Ar0[1];
        v2f a1; a1.x = Ar1[0]; a1.y = Ar1[1];
        v2f a2; a2.x = Ar2[0]; a2.y = Ar2[1];
        v2f a3; a3.x = Ar3[0]; a3.y = Ar3[1];
        c0 = WMMA_F32(a0, b, c0);
        c1 = WMMA_F32(a1, b, c1);
        c2 = WMMA_F32(a2, b, c2);
        c3 = WMMA_F32(a3, b, c3);
        Ar0 += 4; Ar1 += 4; Ar2 += 4; Ar3 += 4;
        Bc  += (size_t)4 * ldb;
    }

    const int mb = (lane < 16) ? 0 : 8;
    float* Cr = C + (size_t)(m0 + mb) * ldc + n0 + mn;
#pragma unroll
    for (int j = 0; j < 8; ++j) Cr[(size_t)j * ldc] = c0[j];
    Cr += (size_t)16 * ldc;
#pragma unroll
    for (int j = 0; j < 8; ++j) Cr[(size_t)j * ldc] = c1[j];
    Cr += (size_t)16 * ldc;
#pragma unroll
    for (int j = 0; j < 8; ++j) Cr[(size_t)j * ldc] = c2[j];
    Cr += (size_t)16 * ldc;
#pragma unroll
    for (int j = 0; j < 8; ++j) Cr[(size_t)j * ldc] = c3[j];
}

// ---------------------------------------------------------------------------
// Generic row-major GEMM, one wave per 16x16 C tile (for the tiny hV GEMM).
// ---------------------------------------------------------------------------
__global__ void gemm_wmma_k(const float* __restrict__ A,
                            const float* __restrict__ Bm,
                            float* __restrict__ C,
                            int M, int N, int K, int lda, int ldb, int ldc) {
    const int wid  = (blockIdx.x * blockDim.x + threadIdx.x) >> 5;
    const int lane = threadIdx.x & 31;
    const int ntiles = N >> 4;
    const int nt = wid % ntiles;
    const int mt = wid / ntiles;
    if (mt * 16 >= M) return;                    // wave-uniform guard
    const int n0 = nt << 4, m0 = mt << 4;
    const int mn = lane & 15;
    const int kb = (lane < 16) ? 0 : 2;

    const float* Ar = A  + (size_t)(m0 + mn) * lda + kb;
    const float* Bc = Bm + (size_t)kb * ldb + n0 + mn;

    v8f c = {0.f,0.f,0.f,0.f,0.f,0.f,0.f,0.f};
    for (int k0 = 0; k0 < K; k0 += 4) {
        v2f a, b;
        a.x = Ar[0];
        a.y = Ar[1];
        b.x = Bc[0];
        b.y = Bc[ldb];
        c = WMMA_F32(a, b, c);
        Ar += 4;
        Bc += (size_t)4 * ldb;
    }
    const int mb = (lane < 16) ? 0 : 8;
    float* Cr = C + (size_t)(m0 + mb) * ldc + n0 + mn;
#pragma unroll
    for (int j = 0; j < 8; ++j) Cr[(size_t)j * ldc] = c[j];
}

// ---------------------------------------------------------------------------
// LSTM gates GEMM: A = concat(x_t, h_prev) (32 x 256), B = w_lstm (256 x 512).
// Concat handled by per-lane address select (the K pair never straddles the
// 128 boundary since kb is even).  64 waves (2 m-tiles x 32 n-tiles).
// ---------------------------------------------------------------------------
__global__ void gates_wmma_k(const float* __restrict__ xemb,
                             const float* __restrict__ hprev,
                             const float* __restrict__ wl,
                             float* __restrict__ gates, int t) {
    const int wid  = (blockIdx.x * blockDim.x + threadIdx.x) >> 5;
    const int lane = threadIdx.x & 31;
    const int nt = wid & 31;
    const int mt = wid >> 5;
    const int n0 = nt << 4, m0 = mt << 4;
    const int mn = lane & 15;
    const int kb = (lane < 16) ? 0 : 2;
    const int m  = m0 + mn;                      // batch row

    const float* xr = xemb + ((size_t)m * L_ + t) * H_;   // x_t row
    const float* hr = hprev + (size_t)m * H_;             // h row
    const float* Bc = wl + (size_t)kb * (4 * H_) + n0 + mn;

    v8f c = {0.f,0.f,0.f,0.f,0.f,0.f,0.f,0.f};
#pragma unroll 4
    for (int k0 = 0; k0 < 2 * H_; k0 += 4) {
        const int ka = k0 + kb;
        const float* pa = (ka < H_) ? (xr + ka) : (hr + (ka - H_));
        v2f a, b;
        a.x = pa[0];
        a.y = pa[1];
        b.x = Bc[0];
        b.y = Bc[4 * H_];
        c = WMMA_F32(a, b, c);
        Bc += (size_t)4 * (4 * H_);
    }
    const int mb = (lane < 16) ? 0 : 8;
    float* Cr = gates + (size_t)(m0 + mb) * (4 * H_) + n0 + mn;
#pragma unroll
    for (int j = 0; j < 8; ++j) Cr[(size_t)j * (4 * H_)] = c[j];
}

// ---------------------------------------------------------------------------
// LSTM pointwise: c_new = sig(f)*c + sig(i)*tanh(g); h_new = sig(o)*tanh(c_new)
// ---------------------------------------------------------------------------
__device__ __forceinline__ float sigf(float x) { return 1.f / (1.f + expf(-x)); }

__global__ void lstm_pw_k(const float* __restrict__ gates,
                          float* __restrict__ cbuf,
                          float* __restrict__ hnew) {
    const int idx = blockIdx.x * blockDim.x + threadIdx.x;   // 0..4095
    const int b = idx >> 7, hh = idx & 127;
    const float gi = gates[(size_t)b * 512 + hh];
    const float gf = gates[(size_t)b * 512 + 128 + hh];
    const float go = gates[(size_t)b * 512 + 256 + hh];
    const float gg = gates[(size_t)b * 512 + 384 + hh];
    const float cn = sigf(gf) * cbuf[idx] + sigf(gi) * tanhf(gg);
    cbuf[idx] = cn;
    hnew[idx] = sigf(go) * tanhf(cn);
}

// ---------------------------------------------------------------------------
// Attention step: one block (256 thr = 8 waves) per batch element.
//  scores[e] = sum_h tanh(enc_proj[b,e,h] + hV[b,h]); softmax over e;
//  context[b,h] = sum_e alpha[e]*enc_output[b,e,h];  h_next = context.
// ---------------------------------------------------------------------------
__global__ void attn_step_k(const float* __restrict__ enc_proj,
                            const float* __restrict__ enc_out,
                            const float* __restrict__ hV,
                            float* __restrict__ hbuf,
                            float* __restrict__ hs,
                            float* __restrict__ alphas, int t) {
    __shared__ float s_hv[H_];
    __shared__ float s_sc[E_];
    __shared__ float s_red[256];

    const int b = blockIdx.x;
    const int tid = threadIdx.x;
    const int lane = tid & 31, wave = tid >> 5;   // 8 waves

    if (tid < H_) s_hv[tid] = hV[(size_t)b * H_ + tid];
    __syncthreads();

    // ---- scores: each wave owns 128 rows; coalesced 4-float strip per lane
    const float* ep = enc_proj + (size_t)b * E_ * H_;
    const int h0 = lane * 4;
    for (int r = 0; r < 128; ++r) {
        const int e = wave * 128 + r;
        const float* row = ep + (size_t)e * H_;
        float acc = 0.f;
#pragma unroll
        for (int j = 0; j < 4; ++j) acc += tanhf(row[h0 + j] + s_hv[h0 + j]);
        for (int off = 16; off > 0; off >>= 1) acc += __shfl_xor(acc, off, 32);
        if (lane == 0) s_sc[e] = acc;
    }
    __syncthreads();

    // ---- softmax over 1024 scores
    float lmax = -1e30f;
    for (int e = tid; e < E_; e += 256) lmax = fmaxf(lmax, s_sc[e]);
    s_red[tid] = lmax;
    __syncthreads();
    for (int s = 128; s > 0; s >>= 1) {
        if (tid < s) s_red[tid] = fmaxf(s_red[tid], s_red[tid + s]);
        __syncthreads();
    }
    const float gmax = s_red[0];
    __syncthreads();

    float lsum = 0.f;
    for (int e = tid; e < E_; e += 256) {
        const float ex = expf(s_sc[e] - gmax);
        s_sc[e] = ex;
        lsum += ex;
    }
    s_red[tid] = lsum;
    __syncthreads();
    for (int s = 128; s > 0; s >>= 1) {
        if (tid < s) s_red[tid] += s_red[tid + s];
        __syncthreads();
    }
    const float inv = 1.f / s_red[0];
    __syncthreads();

    for (int e = tid; e < E_; e += 256) {
        const float al = s_sc[e] * inv;
        s_sc[e] = al;
        alphas[((size_t)b * L_ + t) * E_ + e] = al;
    }
    __syncthreads();

    // ---- context: 128 h-lanes x 2 e-halves, coalesced over h
    const int hh = tid & 127, half = tid >> 7;
    const float* eo = enc_out + (size_t)b * E_ * H_;
    float acc = 0.f;
    for (int e = half * 512; e < half * 512 + 512; ++e)
        acc += s_sc[e] * eo[(size_t)e * H_ + hh];
    if (half == 1) s_red[hh] = acc;
    __syncthreads();
    if (half == 0) {
        const float ctx = acc + s_red[hh];
        hbuf[(size_t)b * H_ + hh] = ctx;
        hs[((size_t)b * L_ + t) * H_ + hh] = ctx;
    }
}

// ---------------------------------------------------------------------------
// Small helpers: embedding gather, state init, final state copy-out.
// ---------------------------------------------------------------------------
__global__ void embed_k(const int* __restrict__ sent,
                        const float* __restrict__ w_emb,
                        float* __restrict__ xemb) {
    const int idx = blockIdx.x * blockDim.x + threadIdx.x;   // B*L*H
    const int hh = idx & 127, bl = idx >> 7;
    xemb[idx] = w_emb[(size_t)sent[bl] * H_ + hh];
}

__global__ void initch_k(const float* __restrict__ ic, const float* __restrict__ ih,
                         float* __restrict__ c, float* __restrict__ h) {
    const int i = blockIdx.x * blockDim.x + threadIdx.x;
    if (i < B_ * H_) { c[i] = ic[i]; h[i] = ih[i]; }
}

__global__ void fincopy_k(const float* __restrict__ c, const float* __restrict__ h,
                          float* __restrict__ oc, float* __restrict__ oh) {
    const int i = blockIdx.x * blockDim.x + threadIdx.x;
    if (i < B_ * H_) { oc[i] = c[i]; oh[i] = h[i]; }
}

// ---------------------------------------------------------------------------
extern "C" void kernel_launch(void* const* d_in, const int* in_sizes, int n_in,
                              void* d_out, int out_size, void* d_ws, size_t ws_size,
                              hipStream_t stream) {
    const int*   sent    = (const int*)  d_in[0];
    const float* init_c  = (const float*)d_in[1];
    const float* init_h  = (const float*)d_in[2];
    const float* enc_out = (const float*)d_in[3];
    const float* w_emb   = (const float*)d_in[4];
    const float* w_lstm  = (const float*)d_in[5];
    const float* w_soft  = (const float*)d_in[6];
    const float* V       = (const float*)d_in[7];

    // Output layout: logits | c_fin | h_fin | alphas
    float* out    = (float*)d_out;
    float* logits = out;
    float* out_c  = out + (size_t)B_ * L_ * VOC_;
    float* out_h  = out_c + (size_t)B_ * H_;
    float* alphas = out_h + (size_t)B_ * H_;

    // Workspace layout (floats)
    float* ws = (float*)d_ws;
    float* xemb = ws;                                   // B*L*H   = 262144
    float* encp = xemb + (size_t)B_ * L_ * H_;          // B*E*H   = 4194304
    float* cbuf = encp + (size_t)B_ * E_ * H_;          // B*H
    float* hbuf = cbuf + (size_t)B_ * H_;               // B*H
    float* hnew = hbuf + (size_t)B_ * H_;               // B*H
    float* hV   = hnew + (size_t)B_ * H_;               // B*H
    float* gat  = hV   + (size_t)B_ * H_;               // B*4H
    float* hs   = gat  + (size_t)B_ * 4 * H_;           // B*L*H

    // Embedding gather + state init
    embed_k<<<(B_ * L_ * H_) / 256, 256, 0, stream>>>(sent, w_emb, xemb);
    initch_k<<<16, 256, 0, stream>>>(init_c, init_h, cbuf, hbuf);

    // enc_proj = enc_output @ V_e : (32768 x 128) @ (128 x 128)
    // waves = (32768/64) * (128/16) = 4096 -> 1024 blocks of 128 threads
    gemm_wmma_mb4_k<<<1024, 128, 0, stream>>>(enc_out, V, encp,
                                              B_ * E_, H_, H_, H_, H_);

    for (int t = 0; t < L_; ++t) {
        // gates = [x_t, h] @ w_lstm : 64 waves
        gates_wmma_k<<<16, 128, 0, stream>>>(xemb, hbuf, w_lstm, gat, t);
        // LSTM pointwise
        lstm_pw_k<<<16, 256, 0, stream>>>(gat, cbuf, hnew);
        // hV = h_new @ V_h : (32 x 128) @ (128 x 128) : 16 waves
        gemm_wmma_k<<<4, 128, 0, stream>>>(hnew, V + (size_t)H_ * H_, hV,
                                           B_, H_, H_, H_, H_, H_);
        // attention + softmax + context (updates hbuf, writes hs & alphas)
        attn_step_k<<<B_, 256, 0, stream>>>(encp, enc_out, hV, hbuf, hs, alphas, t);
    }

    // logits = hs @ w_soft : (2048 x 128) @ (128 x 32000)
    // waves = (2048/64) * (32000/16) = 64000 -> 16000 blocks of 128 threads
    gemm_wmma_mb4_k<<<16000, 128, 0, stream>>>(hs, w_soft, logits,
                                               B_ * L_, VOC_, H_, VOC_, VOC_);

    fincopy_k<<<16, 256, 0, stream>>>(cbuf, hbuf, out_c, out_h);
}